// HRATransform_44641890074624
// MI455X (gfx1250) — compile-verified
//
#include <hip/hip_runtime.h>
#include <hip/hip_bf16.h>

// HRA (Householder Reflection Adaptation) via compact-WY:
//   out = x - (x*Uhat) * T * Uhat^T        (T: 8x8 upper triangular, diag 2)
// Kernels:
//   1) hra_setup : normalize U, build T, transpose/negate U, zero P
//   2) hra_proj  : P = x * Uhat          (WMMA f32 16x16x4, K-split + f32 atomics)
//   3) hra_apply : out = x + (P*T)*(-Uhat^T)   (WMMA with x as the C operand)

typedef __attribute__((ext_vector_type(2))) float v2f;
typedef __attribute__((ext_vector_type(8))) float v8f;

#define DDIM   4096
#define RANK   8
#define NROWS  8192          // 4 * 2048
#define KSPLIT 4
#define KSLICE (DDIM / KSPLIT)   // 1024

// Workspace layout (float offsets):
//   Uhat  : [0,     32768)   4096 x 8, row-major (d-major)
//   negUt : [32768, 65536)   8 x 4096, row-major (-Uhat^T)
//   T     : [65536, 65600)   8 x 8 row-major
//   P     : [65600, 131136)  8192 x 8
#define WS_UHAT  0
#define WS_NEGUT 32768
#define WS_T     65536
#define WS_P     65600

// ---------------------------------------------------------------- setup ----
__global__ void __launch_bounds__(256) hra_setup(const float* __restrict__ u,
                                                 float* __restrict__ ws) {
    __shared__ float sG[64];   // Gram matrix of raw u columns
    __shared__ float sRN[8];   // reciprocal norms
    const int tid = threadIdx.x;

    if (tid < 64) sG[tid] = 0.0f;
    __syncthreads();

    // per-thread partial Gram over strided d
    float g[64];
    #pragma unroll
    for (int i = 0; i < 64; ++i) g[i] = 0.0f;
    for (int d = tid; d < DDIM; d += 256) {
        const float4* p = (const float4*)(u + (size_t)d * RANK);
        float4 a = p[0], b = p[1];
        float uv[8] = {a.x, a.y, a.z, a.w, b.x, b.y, b.z, b.w};
        #pragma unroll
        for (int i = 0; i < 8; ++i)
            #pragma unroll
            for (int j = 0; j < 8; ++j)
                g[i * 8 + j] += uv[i] * uv[j];
    }
    #pragma unroll
    for (int i = 0; i < 64; ++i) atomicAdd(&sG[i], g[i]);   // ds_add_f32
    __syncthreads();

    if (tid == 0) {
        float rn[8];
        #pragma unroll
        for (int i = 0; i < 8; ++i) rn[i] = 1.0f / sqrtf(sG[i * 8 + i]);
        #pragma unroll
        for (int i = 0; i < 8; ++i) sRN[i] = rn[i];
        // normalized Gram
        float Gh[8][8];
        for (int i = 0; i < 8; ++i)
            for (int j = 0; j < 8; ++j)
                Gh[i][j] = sG[i * 8 + j] * rn[i] * rn[j];
        // compact-WY T: Q = H1..H8 = I - U T U^T
        float Tm[8][8];
        for (int i = 0; i < 8; ++i)
            for (int j = 0; j < 8; ++j) Tm[i][j] = 0.0f;
        Tm[0][0] = 2.0f;
        for (int k = 1; k < 8; ++k) {
            for (int j = 0; j < k; ++j) {
                float s = 0.0f;
                for (int m = j; m < k; ++m) s += Tm[j][m] * Gh[m][k];
                Tm[j][k] = -2.0f * s;
            }
            Tm[k][k] = 2.0f;
        }
        float* Tws = ws + WS_T;
        for (int i = 0; i < 8; ++i)
            for (int j = 0; j < 8; ++j) Tws[i * 8 + j] = Tm[i][j];
    }
    __syncthreads();

    float rn[8];
    #pragma unroll
    for (int i = 0; i < 8; ++i) rn[i] = sRN[i];
    float* Uhat  = ws + WS_UHAT;
    float* negUt = ws + WS_NEGUT;
    for (int d = tid; d < DDIM; d += 256) {
        #pragma unroll
        for (int i = 0; i < 8; ++i) {
            float uh = u[(size_t)d * RANK + i] * rn[i];
            Uhat[(size_t)d * RANK + i]  = uh;
            negUt[(size_t)i * DDIM + d] = -uh;
        }
    }
    // zero the P accumulator (atomically accumulated by hra_proj each call)
    float* P = ws + WS_P;
    for (int idx = tid; idx < NROWS * RANK; idx += 256) P[idx] = 0.0f;
}

// ----------------------------------------------------- P = x * Uhat -------
// grid: 64 rowBlocks x KSPLIT;  block: 256 threads = 8 waves
// wave w of a block handles rows [ (rowBlock*8+w)*16 , +16 ), K-range of its ktile
__global__ void __launch_bounds__(256) hra_proj(const float* __restrict__ x,
                                                const float* __restrict__ ws,
                                                float* __restrict__ P) {
    __shared__ float sU[KSLICE * RANK];   // 32 KB slice of Uhat for this ktile
    const float* Uhat = ws + WS_UHAT;

    const int ktile    = blockIdx.x & (KSPLIT - 1);
    const int rowBlock = blockIdx.x >> 2;
    const int k0       = ktile * KSLICE;

    {   // cooperative LDS stage of the Uhat K-slice
        const float4* src = (const float4*)(Uhat + (size_t)k0 * RANK);
        float4* dst = (float4*)sU;
        for (int i = threadIdx.x; i < (KSLICE * RANK) / 4; i += 256) dst[i] = src[i];
    }
    __syncthreads();

    const int wave = threadIdx.x >> 5;
    const int lane = threadIdx.x & 31;
    const int half = lane >> 4;     // 0: K+0/K+1 , 1: K+2/K+3  (A/B layout)
    const int m    = lane & 15;     // A row / B col

    const int rowTile = rowBlock * 8 + wave;
    const float* xrow = x + (size_t)(rowTile * 16 + m) * DDIM + k0 + 2 * half;
    const float bmask = (m < RANK) ? 1.0f : 0.0f;   // mask, not a branch: EXEC stays all-1s
    const int   bcol  = m & (RANK - 1);

    v8f acc = {};
    #pragma unroll 4
    for (int kk = 0; kk < KSLICE; kk += 4) {
        float2 av = *(const float2*)(xrow + kk);
        v2f a; a.x = av.x; a.y = av.y;
        v2f b;
        b.x = sU[(kk + 2 * half)     * RANK + bcol] * bmask;
        b.y = sU[(kk + 2 * half + 1) * RANK + bcol] * bmask;
        acc = __builtin_amdgcn_wmma_f32_16x16x4_f32(false, a, false, b,
                                                    (short)0, acc, false, false);
    }

    // D layout: vgpr g, lanes 0-15 -> row rowTile*16+g, lanes 16-31 -> +8; col = m
    if (m < RANK) {
        const int rbase = rowTile * 16 + 8 * half;
        #pragma unroll
        for (int g = 0; g < 8; ++g)
            atomicAdd(&P[(size_t)(rbase + g) * RANK + m], acc[g]);  // global_atomic_add_f32
    }
}

// --------------------------------------- out = x + (P*T) * (-Uhat^T) ------
// grid: 512 row tiles; block: 256 threads = 8 waves; wave w covers d-range [w*512, +512)
__global__ void __launch_bounds__(256) hra_apply(const float* __restrict__ x,
                                                 const float* __restrict__ ws,
                                                 float* __restrict__ out) {
    __shared__ float sW[16 * RANK];   // W = P*T for this 16-row tile
    __shared__ float sT[64];

    const float* negUt = ws + WS_NEGUT;
    const float* Tws   = ws + WS_T;
    const float* P     = ws + WS_P;

    const int rowBase = blockIdx.x * 16;
    const int tid = threadIdx.x;

    if (tid < 64) sT[tid] = Tws[tid];
    __syncthreads();
    if (tid < 128) {                 // W[r][i] = sum_j P[r][j] * T[j][i]
        const int r = tid >> 3, i = tid & 7;
        const float* prow = P + (size_t)(rowBase + r) * RANK;
        float s = 0.0f;
        #pragma unroll
        for (int j = 0; j < 8; ++j) s += prow[j] * sT[j * 8 + i];
        sW[r * RANK + i] = s;
    }
    __syncthreads();

    const int wave = tid >> 5;
    const int lane = tid & 31;
    const int half = lane >> 4;
    const int m    = lane & 15;

    float wrow[8];                   // W row this lane feeds into the A operand
    #pragma unroll
    for (int i = 0; i < 8; ++i) wrow[i] = sW[m * RANK + i];

    const int d0base = wave * (DDIM / 8);
    for (int dt = 0; dt < DDIM / 8; dt += 16) {
        const int col = d0base + dt + m;

        // C operand = x tile (the WMMA accumulate gives out = x + W*(-U^T))
        v8f c;
        const float* xc = x + (size_t)(rowBase + 8 * half) * DDIM + col;
        #pragma unroll
        for (int g = 0; g < 8; ++g) c[g] = xc[(size_t)g * DDIM];

        // K block 0 (reflections 0..3)
        v2f a0, b0;
        a0.x = wrow[2 * half];     a0.y = wrow[2 * half + 1];
        b0.x = negUt[(size_t)(2 * half)     * DDIM + col];
        b0.y = negUt[(size_t)(2 * half + 1) * DDIM + col];
        v8f d = __builtin_amdgcn_wmma_f32_16x16x4_f32(false, a0, false, b0,
                                                      (short)0, c, false, false);
        // K block 1 (reflections 4..7)
        v2f a1, b1;
        a1.x = wrow[4 + 2 * half]; a1.y = wrow[4 + 2 * half + 1];
        b1.x = negUt[(size_t)(4 + 2 * half)     * DDIM + col];
        b1.y = negUt[(size_t)(4 + 2 * half + 1) * DDIM + col];
        d = __builtin_amdgcn_wmma_f32_16x16x4_f32(false, a1, false, b1,
                                                  (short)0, d, false, false);

        float* oc = out + (size_t)(rowBase + 8 * half) * DDIM + col;
        #pragma unroll
        for (int g = 0; g < 8; ++g) oc[(size_t)g * DDIM] = d[g];
    }
}

// ------------------------------------------------------------------ host --
extern "C" void kernel_launch(void* const* d_in, const int* in_sizes, int n_in,
                              void* d_out, int out_size, void* d_ws, size_t ws_size,
                              hipStream_t stream) {
    const float* x = (const float*)d_in[0];   // (4, 2048, 4096) f32
    const float* u = (const float*)d_in[1];   // (4096, 8) f32
    float* out = (float*)d_out;
    float* ws  = (float*)d_ws;                // needs ~513 KB

    hipLaunchKernelGGL(hra_setup, dim3(1), dim3(256), 0, stream, u, ws);
    hipLaunchKernelGGL(hra_proj,  dim3((NROWS / 16 / 8) * KSPLIT), dim3(256), 0, stream,
                       x, ws, ws + WS_P);
    hipLaunchKernelGGL(hra_apply, dim3(NROWS / 16), dim3(256), 0, stream,
                       x, ws, out);
}